// Rank1Linear_18691697672413
// MI455X (gfx1250) — compile-verified
//
#include <hip/hip_runtime.h>

typedef float v2f __attribute__((ext_vector_type(2)));
typedef float v4f __attribute__((ext_vector_type(4)));
typedef float v8f __attribute__((ext_vector_type(8)));

#define BLOCK 256
#define NWAVES (BLOCK / 32)

// Full wave32 sum via V_WMMA_F32_16X16X4_F32 (exact f32 FMA accumulation).
//
// Step 1: A = (p, 0) per the documented 16x4 f32 A layout
//           (lane L: M = L&15; VGPR0 -> K = 2*(L>>4), VGPR1 -> K = 2*(L>>4)+1)
//         B = all-ones  =>  D1[m][n] = p[m] + p[m+16]          (fold halves)
//         D layout (documented): d1[r] at lane L = q[r + 8*(L>>4)],
//           where q[M] = p[M] + p[M+16].
// Step 2: A = all-ones, B = (d1[2j], d1[2j+1]), j = 0..3, accumulating:
//         each WMMA adds q[2j]+q[2j+1]+q[2j+8]+q[2j+9] to every element
//         (any B layout sums the same 4 K-slots), so after 4 WMMAs every
//         element of acc equals sum_{L=0..31} p[L] -> uniform broadcast.
__device__ __forceinline__ float wave_sum32_wmma(float p) {
  v2f a;
  a[0] = p;
  a[1] = 0.0f;
  v2f ones;
  ones[0] = 1.0f;
  ones[1] = 1.0f;

  v8f zero_c = {};
  v8f d1 = __builtin_amdgcn_wmma_f32_16x16x4_f32(
      /*neg_a=*/false, a, /*neg_b=*/false, ones,
      /*c_mod=*/(short)0, zero_c, /*reuse_a=*/false, /*reuse_b=*/false);

  v8f acc = {};
  v2f b;
#pragma unroll
  for (int j = 0; j < 4; ++j) {
    b[0] = d1[2 * j];
    b[1] = d1[2 * j + 1];
    acc = __builtin_amdgcn_wmma_f32_16x16x4_f32(
        false, ones, false, b, (short)0, acc, false, false);
  }
  return acc[0];  // identical in every lane
}

// Shared tail: WMMA wave reduction + 8-wave LDS fan-in. Uniform result.
__device__ __forceinline__ float block_rowsum(float p, int tid) {
  float wsum = wave_sum32_wmma(p);  // EXEC all-ones here (uniform flow)

  __shared__ float lds_part[NWAVES];
  if ((tid & 31) == 0) lds_part[tid >> 5] = wsum;
  __syncthreads();

  float rowsum = 0.0f;
#pragma unroll
  for (int i = 0; i < NWAVES; ++i) rowsum += lds_part[i];
  return rowsum;
}

// Specialized: one block per row, trip counts known at compile time so all
// loops fully unroll into straight-line vmem + WMMA code.
//  Phase 1: KITERS coalesced non-temporal b128 loads of the row per thread.
//  Phase 2: NITERS b128 w loads (regular-temporal: 16 KB, L2-resident) and
//           NITERS non-temporal b128 stores of rowsum * w.
template <int KITERS, int NITERS>
__global__ __launch_bounds__(BLOCK) void Rank1Linear_kernel_tmpl(
    const float* __restrict__ x, const float* __restrict__ w,
    float* __restrict__ out) {
  const int row = blockIdx.x;
  const int tid = threadIdx.x;

  const v4f* __restrict__ xrow =
      reinterpret_cast<const v4f*>(x) + (size_t)row * (size_t)(KITERS * BLOCK);

  float p = 0.0f;
#pragma unroll
  for (int it = 0; it < KITERS; ++it) {
    v4f v = __builtin_nontemporal_load(&xrow[tid + it * BLOCK]);
    p += (v[0] + v[1]) + (v[2] + v[3]);
  }

  const float rowsum = block_rowsum(p, tid);

  const v4f* __restrict__ wv = reinterpret_cast<const v4f*>(w);
  v4f* __restrict__ orow =
      reinterpret_cast<v4f*>(out) + (size_t)row * (size_t)(NITERS * BLOCK);

#pragma unroll
  for (int it = 0; it < NITERS; ++it) {
    const int i = tid + it * BLOCK;
    v4f wl = wv[i];
    v4f o;
    o[0] = rowsum * wl[0];
    o[1] = rowsum * wl[1];
    o[2] = rowsum * wl[2];
    o[3] = rowsum * wl[3];
    __builtin_nontemporal_store(o, &orow[i]);
  }
}

// Generic fallback: runtime trip counts; K, N must be
// multiples of 4. K loop: uniform bound -> scalar branch; the N loop's
// bound is per-lane (tid-strided), so it is exec-masked.
__global__ __launch_bounds__(BLOCK) void Rank1Linear_kernel_gen(
    const float* __restrict__ x, const float* __restrict__ w,
    float* __restrict__ out, int kvec, int nvec) {
  const int row = blockIdx.x;
  const int tid = threadIdx.x;

  const v4f* __restrict__ xrow =
      reinterpret_cast<const v4f*>(x) + (size_t)row * (size_t)kvec;

  float p = 0.0f;
  const int kiters = kvec >> 8;  // kvec / BLOCK
  for (int it = 0; it < kiters; ++it) {
    v4f v = __builtin_nontemporal_load(&xrow[tid + (it << 8)]);
    p += (v[0] + v[1]) + (v[2] + v[3]);
  }
  {  // tail (uniformly sized; per-lane guard only here)
    const int i = (kiters << 8) + tid;
    if (i < kvec) {
      v4f v = __builtin_nontemporal_load(&xrow[i]);
      p += (v[0] + v[1]) + (v[2] + v[3]);
    }
  }

  const float rowsum = block_rowsum(p, tid);

  const v4f* __restrict__ wv = reinterpret_cast<const v4f*>(w);
  v4f* __restrict__ orow =
      reinterpret_cast<v4f*>(out) + (size_t)row * (size_t)nvec;

  for (int i = tid; i < nvec; i += BLOCK) {
    v4f wl = wv[i];
    v4f o;
    o[0] = rowsum * wl[0];
    o[1] = rowsum * wl[1];
    o[2] = rowsum * wl[2];
    o[3] = rowsum * wl[3];
    __builtin_nontemporal_store(o, &orow[i]);
  }
}

extern "C" void kernel_launch(void* const* d_in, const int* in_sizes, int n_in,
                              void* d_out, int out_size, void* d_ws,
                              size_t ws_size, hipStream_t stream) {
  const float* x = (const float*)d_in[0];
  const float* w = (const float*)d_in[1];
  float* out = (float*)d_out;

  const int N = in_sizes[1];          // out_features (w has shape (1, N))
  const int batch = out_size / N;     // rows of x / out
  const int K = in_sizes[0] / batch;  // in_features

  dim3 grid(batch);
  dim3 block(BLOCK);

  if (K == 4096 && N == 4096) {
    // 4096/4 = 1024 float4 per row; 1024/256 = 4 per thread each phase.
    hipLaunchKernelGGL((Rank1Linear_kernel_tmpl<4, 4>), grid, block, 0, stream,
                       x, w, out);
  } else {
    hipLaunchKernelGGL(Rank1Linear_kernel_gen, grid, block, 0, stream,
                       x, w, out, K >> 2, N >> 2);
  }
}